// TopologyEncoderSignOnly_50800873177282
// MI455X (gfx1250) — compile-verified
//
#include <hip/hip_runtime.h>
#include <hip/hip_bf16.h>

// ---------------------------------------------------------------------------
// Signed-GCN (pos/neg branch) encoder for MI455X (gfx1250, wave32).
// Strategy:
//   deg/edge-norm precompute (once) ->
//   per layer per branch:  t = selfn*h  ;  t[dst] += en[e]*h[src] (atomics) ;
//                          t = t @ W + b   (in-place WMMA f32 16x16x4 GEMM)
//   h = relu(t_pos) - relu(t_neg) ; then mean-pool per graph + LayerNorm.
// ---------------------------------------------------------------------------

#define D_DIM 128
#define EPS_LN 1e-5f

typedef float v2f __attribute__((ext_vector_type(2)));
typedef float v8f __attribute__((ext_vector_type(8)));

// ---------------- degree / normalization precompute ----------------

__global__ void init_deg(float* __restrict__ dp, float* __restrict__ dn, int N) {
    int i = blockIdx.x * blockDim.x + threadIdx.x;
    if (i < N) { dp[i] = 1.0f; dn[i] = 1.0f; }   // self-loop contributes 1
}

__global__ void count_deg(const float* __restrict__ ew, const int* __restrict__ dst,
                          float* __restrict__ dp, float* __restrict__ dn, int E) {
    int e = blockIdx.x * blockDim.x + threadIdx.x;
    if (e >= E) return;
    float w = ew[e];
    int d = dst[e];
    if (w > 0.0f)      atomicAdd(&dp[d], 1.0f);
    else if (w < 0.0f) atomicAdd(&dn[d], 1.0f);
}

// deg -> dinv_sqrt (in place) and self_norm = 1/deg
__global__ void finalize_deg(float* __restrict__ dp, float* __restrict__ dn,
                             float* __restrict__ sp, float* __restrict__ sn, int N) {
    int i = blockIdx.x * blockDim.x + threadIdx.x;
    if (i >= N) return;
    float a = dp[i], b = dn[i];
    sp[i] = 1.0f / a;  dp[i] = rsqrtf(a);
    sn[i] = 1.0f / b;  dn[i] = rsqrtf(b);
}

__global__ void edge_norms(const float* __restrict__ ew,
                           const int* __restrict__ src, const int* __restrict__ dst,
                           const float* __restrict__ dp, const float* __restrict__ dn,
                           float* __restrict__ enp, float* __restrict__ enn, int E) {
    int e = blockIdx.x * blockDim.x + threadIdx.x;
    if (e >= E) return;
    float w = ew[e];
    int s = src[e], d = dst[e];
    enp[e] = (w > 0.0f) ? dp[s] * dp[d] : 0.0f;
    enn[e] = (w < 0.0f) ? dn[s] * dn[d] : 0.0f;
}

// ---------------- per-layer node/edge kernels ----------------

// t[i,:] = selfn[i] * h[i,:]   (also serves as accumulator init)
__global__ void init_t(float* __restrict__ t, const float* __restrict__ h,
                       const float* __restrict__ selfn, int N) {
    long long i = (long long)blockIdx.x * blockDim.x + threadIdx.x;   // float4 index
    long long n4 = (long long)N * (D_DIM / 4);
    if (i >= n4) return;
    int node = (int)(i >> 5);                         // 32 float4 per row
    float s = selfn[node];
    float4 v = ((const float4*)h)[i];
    v.x *= s; v.y *= s; v.z *= s; v.w *= s;
    ((float4*)t)[i] = v;
}

// one wave per edge: t[dst,:] += en[e] * h[src,:]
__global__ void edge_agg(const float* __restrict__ h, const float* __restrict__ en,
                         const int* __restrict__ src, const int* __restrict__ dst,
                         float* __restrict__ t, int E) {
    long long gid = (long long)blockIdx.x * blockDim.x + threadIdx.x;
    int e = (int)(gid >> 5);
    int lane = (int)(gid & 31);
    if (e >= E) return;
    float w = en[e];
    if (w == 0.0f) return;                            // masked-out edge: skip traffic
    int s = src[e], d = dst[e];
    const float4 v = *(const float4*)(h + (long long)s * D_DIM + lane * 4);
    float* o = t + (long long)d * D_DIM + lane * 4;
    atomicAdd(o + 0, w * v.x);
    atomicAdd(o + 1, w * v.y);
    atomicAdd(o + 2, w * v.z);
    atomicAdd(o + 3, w * v.w);
}

// ---------------- in-place WMMA GEMM: t = t @ W + b ----------------
// Block = 128 threads (4 waves), handles 16 rows. A tile staged in LDS.
// Each wave owns two 16-column tiles; K walked in 32 x (16x16x4 f32 WMMA).
// In-place is safe: A is fully staged to LDS before any writeback, and each
// block only touches its own 16 rows. No divergence (EXEC all-1s for WMMA).
__global__ void __launch_bounds__(128)
gemm_bias_inplace(float* __restrict__ t, const float* __restrict__ W,
                  const float* __restrict__ bias) {
    __shared__ float As[16 * D_DIM];                  // 8 KB
    const int row0 = blockIdx.x * 16;
    const int tid  = threadIdx.x;

    // stage 16x128 f32 A tile (2048 floats) with b128 loads
    {
        const float4* gsrc = (const float4*)(t + (long long)row0 * D_DIM);
        float4* ldst = (float4*)As;
        for (int i = tid; i < 16 * (D_DIM / 4); i += 128) ldst[i] = gsrc[i];
    }
    __syncthreads();

    const int lane = tid & 31;
    const int wave = tid >> 5;        // 0..3
    const int half = lane >> 4;       // K-pair select (ISA 7.12.2: 32-bit A 16x4)
    const int l16  = lane & 15;       // M for A, N for B/C

    for (int nt = wave * 2; nt < wave * 2 + 2; ++nt) {
        const int col0 = nt * 16;
        v8f c = {};
        for (int k = 0; k < D_DIM; k += 4) {
            const int kk = k + half * 2;
            v2f a;
            a.x = As[l16 * D_DIM + kk];
            a.y = As[l16 * D_DIM + kk + 1];
            v2f b;
            b.x = W[(long long)kk * D_DIM + col0 + l16];
            b.y = W[(long long)(kk + 1) * D_DIM + col0 + l16];
            // 8 args: (neg_a, A, neg_b, B, c_mod, C, reuse_a, reuse_b)
            c = __builtin_amdgcn_wmma_f32_16x16x4_f32(false, a, false, b,
                                                      (short)0, c, false, false);
        }
        const float bv = bias[col0 + l16];
        #pragma unroll
        for (int j = 0; j < 8; ++j) {                 // C: VGPR j -> M = j + 8*half
            const int m = j + half * 8;
            t[(long long)(row0 + m) * D_DIM + col0 + l16] = c[j] + bv;
        }
    }
}

// scalar tail for rows not covered by 16-row WMMA blocks (N % 16)
__global__ void __launch_bounds__(128)
gemm_tail(float* __restrict__ t, const float* __restrict__ W,
          const float* __restrict__ bias, int row_start, int nrows) {
    int row = row_start + blockIdx.x;
    if (row >= nrows) return;
    __shared__ float a[D_DIM];
    int d = threadIdx.x;
    a[d] = t[(long long)row * D_DIM + d];
    __syncthreads();
    float acc = bias[d];
    for (int k = 0; k < D_DIM; ++k) acc += a[k] * W[k * D_DIM + d];
    t[(long long)row * D_DIM + d] = acc;
}

// h = relu(tp) - relu(tn)
__global__ void combine_relu(float* __restrict__ h, const float* __restrict__ tp,
                             const float* __restrict__ tn, int N) {
    long long i = (long long)blockIdx.x * blockDim.x + threadIdx.x;
    long long n4 = (long long)N * (D_DIM / 4);
    if (i >= n4) return;
    float4 a = ((const float4*)tp)[i];
    float4 b = ((const float4*)tn)[i];
    float4 r;
    r.x = fmaxf(a.x, 0.f) - fmaxf(b.x, 0.f);
    r.y = fmaxf(a.y, 0.f) - fmaxf(b.y, 0.f);
    r.z = fmaxf(a.z, 0.f) - fmaxf(b.z, 0.f);
    r.w = fmaxf(a.w, 0.f) - fmaxf(b.w, 0.f);
    ((float4*)h)[i] = r;
}

// ---------------- pooling + layernorm ----------------

__global__ void zero_pool(float* __restrict__ sums, float* __restrict__ counts, int G) {
    int i = blockIdx.x * blockDim.x + threadIdx.x;
    if (i < G * D_DIM) sums[i] = 0.0f;
    if (i < G) counts[i] = 0.0f;
}

__global__ void count_nodes(const int* __restrict__ batch, float* __restrict__ counts, int N) {
    int i = blockIdx.x * blockDim.x + threadIdx.x;
    if (i < N) atomicAdd(&counts[batch[i]], 1.0f);
}

__global__ void pool_sum(const float* __restrict__ h, const int* __restrict__ batch,
                         float* __restrict__ sums, int N) {
    long long gid = (long long)blockIdx.x * blockDim.x + threadIdx.x;
    int i = (int)(gid >> 5);
    int lane = (int)(gid & 31);
    if (i >= N) return;
    int g = batch[i];
    const float4 v = *(const float4*)(h + (long long)i * D_DIM + lane * 4);
    float* o = sums + (long long)g * D_DIM + lane * 4;
    atomicAdd(o + 0, v.x);
    atomicAdd(o + 1, v.y);
    atomicAdd(o + 2, v.z);
    atomicAdd(o + 3, v.w);
}

__global__ void __launch_bounds__(128)
pool_layernorm(const float* __restrict__ sums, const float* __restrict__ counts,
               const float* __restrict__ gamma, const float* __restrict__ beta,
               float* __restrict__ out) {
    __shared__ float red[D_DIM];
    int g = blockIdx.x, d = threadIdx.x;
    float cnt = counts[g];
    if (cnt < 1.0f) cnt = 1.0f;
    float p = sums[g * D_DIM + d] / cnt;

    red[d] = p; __syncthreads();
    for (int s = 64; s > 0; s >>= 1) { if (d < s) red[d] += red[d + s]; __syncthreads(); }
    float mu = red[0] * (1.0f / D_DIM);
    __syncthreads();

    float diff = p - mu;
    red[d] = diff * diff; __syncthreads();
    for (int s = 64; s > 0; s >>= 1) { if (d < s) red[d] += red[d + s]; __syncthreads(); }
    float var = red[0] * (1.0f / D_DIM);

    out[g * D_DIM + d] = diff * rsqrtf(var + EPS_LN) * gamma[d] + beta[d];
}

// ---------------------------------------------------------------------------

static inline int cdiv(long long a, int b) { return (int)((a + b - 1) / b); }

extern "C" void kernel_launch(void* const* d_in, const int* in_sizes, int n_in,
                              void* d_out, int out_size, void* d_ws, size_t ws_size,
                              hipStream_t stream) {
    const float* x     = (const float*)d_in[0];
    const int*   ei    = (const int*)  d_in[1];
    const float* ew    = (const float*)d_in[2];
    const int*   batch = (const int*)  d_in[3];
    const float* Wp    = (const float*)d_in[4];
    const float* bp    = (const float*)d_in[5];
    const float* Wn    = (const float*)d_in[6];
    const float* bn    = (const float*)d_in[7];
    const float* gamma = (const float*)d_in[8];
    const float* beta  = (const float*)d_in[9];

    const int N = in_sizes[0] / D_DIM;
    const int E = in_sizes[2];
    const int L = in_sizes[4] / (D_DIM * D_DIM);
    const int G = out_size / D_DIM;

    const int* src = ei;
    const int* dst = ei + E;

    // workspace carve-up (floats)
    float* ws = (float*)d_ws;
    float* h   = ws;                 ws += (long long)N * D_DIM;
    float* tp  = ws;                 ws += (long long)N * D_DIM;
    float* tn  = ws;                 ws += (long long)N * D_DIM;
    float* enp = ws;                 ws += E;
    float* enn = ws;                 ws += E;
    float* dp  = ws;                 ws += N;   // deg -> dinv_sqrt (pos)
    float* dn  = ws;                 ws += N;   // deg -> dinv_sqrt (neg)
    float* sp  = ws;                 ws += N;   // self_norm pos
    float* sn  = ws;                 ws += N;   // self_norm neg
    float* sums   = ws;              ws += (long long)G * D_DIM;
    float* counts = ws;              ws += G;

    // h = x
    hipMemcpyAsync(h, x, (size_t)N * D_DIM * sizeof(float),
                   hipMemcpyDeviceToDevice, stream);

    // normalization precompute (reused across all layers)
    init_deg<<<cdiv(N, 256), 256, 0, stream>>>(dp, dn, N);
    count_deg<<<cdiv(E, 256), 256, 0, stream>>>(ew, dst, dp, dn, E);
    finalize_deg<<<cdiv(N, 256), 256, 0, stream>>>(dp, dn, sp, sn, N);
    edge_norms<<<cdiv(E, 256), 256, 0, stream>>>(ew, src, dst, dp, dn, enp, enn, E);

    const long long n4 = (long long)N * (D_DIM / 4);
    const int nblk16   = N / 16;
    const int tail     = N - nblk16 * 16;

    for (int l = 0; l < L; ++l) {
        const float* Wpl = Wp + (long long)l * D_DIM * D_DIM;
        const float* Wnl = Wn + (long long)l * D_DIM * D_DIM;
        const float* bpl = bp + (long long)l * D_DIM;
        const float* bnl = bn + (long long)l * D_DIM;

        // positive branch
        init_t<<<cdiv(n4, 256), 256, 0, stream>>>(tp, h, sp, N);
        edge_agg<<<cdiv((long long)E * 32, 256), 256, 0, stream>>>(h, enp, src, dst, tp, E);
        if (nblk16 > 0)
            gemm_bias_inplace<<<nblk16, 128, 0, stream>>>(tp, Wpl, bpl);
        if (tail > 0)
            gemm_tail<<<tail, 128, 0, stream>>>(tp, Wpl, bpl, nblk16 * 16, N);

        // negative branch
        init_t<<<cdiv(n4, 256), 256, 0, stream>>>(tn, h, sn, N);
        edge_agg<<<cdiv((long long)E * 32, 256), 256, 0, stream>>>(h, enn, src, dst, tn, E);
        if (nblk16 > 0)
            gemm_bias_inplace<<<nblk16, 128, 0, stream>>>(tn, Wnl, bnl);
        if (tail > 0)
            gemm_tail<<<tail, 128, 0, stream>>>(tn, Wnl, bnl, nblk16 * 16, N);

        combine_relu<<<cdiv(n4, 256), 256, 0, stream>>>(h, tp, tn, N);
    }

    // global mean pool + layernorm
    zero_pool<<<cdiv((long long)G * D_DIM, 256), 256, 0, stream>>>(sums, counts, G);
    count_nodes<<<cdiv(N, 256), 256, 0, stream>>>(batch, counts, N);
    pool_sum<<<cdiv((long long)N * 32, 256), 256, 0, stream>>>(h, batch, sums, N);
    pool_layernorm<<<G, 128, 0, stream>>>(sums, counts, gamma, beta, (float*)d_out);
}